// QuantizationLayerEventFeature_62027917689289
// MI455X (gfx1250) — compile-verified
//
#include <hip/hip_runtime.h>
#include <hip/hip_bf16.h>
#include <stdint.h>

// ---------------------------------------------------------------------------
// Event feature quantization (EST + VoxGrid + EventFrame + EventCount)
// H=240, W=304, C=9 temporal bins, B=4, N=2M events.
// Scatter/histogram workload: optimized around L2-resident atomics (35 MB
// output fits the 192 MB L2) and CDNA5 Tensor Data Mover async staging of
// the event stream into LDS (TENSORcnt-tracked, double-buffered).
// ---------------------------------------------------------------------------

static constexpr int Hh = 240;
static constexpr int Ww = 304;
static constexpr int Cc = 9;
static constexpr int HW = Hh * Ww;

static constexpr int TILE_EV    = 1024;           // events per TDM tile
static constexpr int TILE_ELEMS = TILE_EV * 5;    // floats per tile (20 KB)

typedef unsigned int u32x4 __attribute__((ext_vector_type(4)));
typedef int          i32x8 __attribute__((ext_vector_type(8)));
typedef int          i32x4 __attribute__((ext_vector_type(4)));

// ---------------------------------------------------------------------------
// TDM: DMA a linear run of fp32 elements from global memory into LDS.
// Tensor DMA Descriptor (D#) per CDNA5 ISA ch.8:
//   group0: count=1, lds_addr, global_addr(57b), type=2
//   group1: workgroup_mask=0 (no cluster), data_size=4B, tensor_dim0 =
//           remaining elements (OOB tail reads return zero), 1-D tile.
// ---------------------------------------------------------------------------
__device__ __forceinline__ void tdm_load_tile(const float* gbase, long elem_off,
                                              unsigned lds_byte, unsigned tile_elems,
                                              long total_elems) {
  unsigned long long ga  = (unsigned long long)(uintptr_t)gbase +
                           (unsigned long long)elem_off * 4ull;
  long remL = total_elems - elem_off;
  unsigned td0 = (remL > 0x7fffffffL) ? 0x7fffffffu : (unsigned)remL;

  u32x4 g0;
  g0.x = 0x1u;                                         // count=1, user mode
  g0.y = lds_byte;                                     // LDS byte address
  g0.z = (unsigned)(ga & 0xffffffffull);               // global_addr[31:0]
  g0.w = ((unsigned)(ga >> 32) & 0x01ffffffu)          // global_addr[56:32]
         | 0x80000000u;                                // type=2 ("image")

  i32x8 g1;
  g1[0] = (int)(2u << 16);                             // data_size=2 (4 bytes)
  g1[1] = (int)((td0 & 0xffffu) << 16);                // tensor_dim0[15:0]
  g1[2] = (int)((td0 >> 16) & 0xffffu) | (1 << 16);    // tensor_dim0[31:16], tensor_dim1=1
  g1[3] = (int)((tile_elems & 0xffffu) << 16);         // tile_dim0
  g1[4] = 0;                                           // tile_dim1=0, tile_dim2=0 (1-D)
  g1[5] = (int)tile_elems;                             // tensor_dim0_stride[31:0]
  g1[6] = 0;
  g1[7] = 0;

  i32x4 z4 = {0, 0, 0, 0};
#if defined(__clang_major__) && (__clang_major__ >= 23)
  i32x8 z8 = {0, 0, 0, 0, 0, 0, 0, 0};
  __builtin_amdgcn_tensor_load_to_lds(g0, g1, z4, z4, z8, 0);
#else
  __builtin_amdgcn_tensor_load_to_lds(g0, g1, z4, z4, 0);
#endif
}

// ---------------------------------------------------------------------------
// Pass 1: max over t (events[:,2]). t > 0, so bitwise uint max == float max.
// wave32 shuffle reduction + one atomic per block.
// ---------------------------------------------------------------------------
__global__ void ef_tmax_kernel(const float* __restrict__ ev, int n,
                               unsigned* __restrict__ wsmax) {
  float m = 0.0f;
  for (int i = blockIdx.x * blockDim.x + threadIdx.x; i < n;
       i += gridDim.x * blockDim.x)
    m = fmaxf(m, ev[5 * i + 2]);

  for (int off = 16; off > 0; off >>= 1) m = fmaxf(m, __shfl_xor(m, off));

  __shared__ float sm[8];
  int lane = threadIdx.x & 31, wid = threadIdx.x >> 5;
  if (lane == 0) sm[wid] = m;
  __syncthreads();
  if (wid == 0) {
    m = (lane < (int)(blockDim.x >> 5)) ? sm[lane] : 0.0f;
    for (int off = 16; off > 0; off >>= 1) m = fmaxf(m, __shfl_xor(m, off));
    if (lane == 0) atomicMax(wsmax, __float_as_uint(m));
  }
}

// ---------------------------------------------------------------------------
// Pass 2: scatter. Wave 0 drives double-buffered TDM tiles into LDS; all 8
// waves consume. Per event: <=2 EST atomics (trilinear support is 2 bins,
// ts==0 quirk -> f==0 contributes nothing), 1 plain VoxGrid store of 1.0f,
// 1 EventCount atomic. Output channels per batch b:
//   [0..17] EST (p*9+bin), [18..26] VoxGrid, [27] EventFrame, [28..29] EC.
// ---------------------------------------------------------------------------
__global__ void ef_scatter_kernel(const float* __restrict__ ev, int N,
                                  const unsigned* __restrict__ wsmax,
                                  float* __restrict__ out,
                                  int num_tiles, int tiles_per_block) {
  __shared__ __align__(16) float smem[2][TILE_ELEMS];

  const float tmax = __uint_as_float(*wsmax);
  const long  total = (long)N * 5;

  int t0 = blockIdx.x * tiles_per_block;
  int t1 = t0 + tiles_per_block;
  if (t1 > num_tiles) t1 = num_tiles;
  if (t0 >= t1) return;

  const bool issuer = (threadIdx.x < 32);  // wave 0 owns the TDM pipeline

  if (issuer)
    tdm_load_tile(ev, (long)t0 * TILE_ELEMS,
                  (unsigned)(uintptr_t)&smem[0][0], TILE_ELEMS, total);

  for (int tk = t0; tk < t1; ++tk) {
    const int buf = (tk - t0) & 1;
    if (issuer) {
      if (tk + 1 < t1) {
        tdm_load_tile(ev, (long)(tk + 1) * TILE_ELEMS,
                      (unsigned)(uintptr_t)&smem[buf ^ 1][0], TILE_ELEMS, total);
        __builtin_amdgcn_s_wait_tensorcnt(1);  // oldest (tile tk) complete
      } else {
        __builtin_amdgcn_s_wait_tensorcnt(0);
      }
    }
    __syncthreads();  // LDS tile tk visible to all waves

    const int ebase = tk * TILE_EV;
    for (int le = threadIdx.x; le < TILE_EV; le += blockDim.x) {
      const int e = ebase + le;
      if (e >= N) break;
      const float* evp = &smem[buf][le * 5];
      const float x = evp[0], y = evp[1], t = evp[2], p = evp[3], b = evp[4];
      const int xi = (int)x, yi = (int)y, pi = (int)p, bi = (int)b;

      const float tn = t / tmax;          // exact <= 1.0 (true division)
      const int pix = yi * Ww + xi;
      float* ob = out + (size_t)bi * 30 * HW;

      // ---- EST: 2-bin trilinear, value tn * w ----
      float u = tn * 8.0f;
      int i0 = (int)u;
      if (i0 > 8) i0 = 8;
      const float f = u - (float)i0;
      if (f > 0.0f) {                      // f==0 -> ts==0 quirk: contributes 0
        atomicAdd(ob + (pi * Cc + i0) * HW + pix, tn * (1.0f - f));
        if (i0 < 8)
          atomicAdd(ob + (pi * Cc + i0 + 1) * HW + pix, tn * f);
      }

      // ---- VoxGrid: bin j with j/9 < tn <= (j+1)/9; binarized -> plain store
      const float v9 = tn * 9.0f;
      int j = (int)v9;
      if ((float)j == v9) j--;             // tn exactly on boundary -> lower bin
      if (j < 0) j = 0;
      if (j > 8) j = 8;
      ob[(18 + j) * HW + pix] = 1.0f;

      // ---- EventCount ----
      atomicAdd(ob + (28 + pi) * HW + pix, 1.0f);
    }
    __syncthreads();  // all waves done with buf before it is DMA-overwritten
  }
}

// ---------------------------------------------------------------------------
// Pass 3: EventFrame = EC(p0) + EC(p1) per pixel.
// ---------------------------------------------------------------------------
__global__ void ef_finalize_kernel(float* __restrict__ out, int B) {
  const int i = blockIdx.x * blockDim.x + threadIdx.x;
  const int tot = B * HW;
  if (i >= tot) return;
  const int b = i / HW, pix = i - b * HW;
  float* ob = out + (size_t)b * 30 * HW;
  ob[27 * HW + pix] = ob[28 * HW + pix] + ob[29 * HW + pix];
}

// ---------------------------------------------------------------------------
extern "C" void kernel_launch(void* const* d_in, const int* in_sizes, int n_in,
                              void* d_out, int out_size, void* d_ws, size_t ws_size,
                              hipStream_t stream) {
  (void)n_in; (void)ws_size;
  const float* events = (const float*)d_in[0];
  const int N = in_sizes[0] / 5;
  float* out = (float*)d_out;
  const int B = out_size / (30 * HW);
  unsigned* wsmax = (unsigned*)d_ws;

  // Zero output (atomics accumulate) and the t-max cell, every call.
  hipMemsetAsync(d_out, 0, (size_t)out_size * sizeof(float), stream);
  hipMemsetAsync(d_ws, 0, sizeof(unsigned), stream);

  ef_tmax_kernel<<<1024, 256, 0, stream>>>(events, N, wsmax);

  const int num_tiles = (N + TILE_EV - 1) / TILE_EV;
  const int tpb = 2;  // 2 tiles/block: full double-buffer overlap, ~977 WGs
  const int blocks = (num_tiles + tpb - 1) / tpb;
  ef_scatter_kernel<<<blocks, 256, 0, stream>>>(events, N, wsmax, out,
                                                num_tiles, tpb);

  const int tot = B * HW;
  ef_finalize_kernel<<<(tot + 255) / 256, 256, 0, stream>>>(out, B);
}